// MoleculeAgent_22024592294374
// MI455X (gfx1250) — compile-verified
//
#include <hip/hip_runtime.h>
#include <hip/hip_bf16.h>
#include <cstdint>

typedef __attribute__((ext_vector_type(2))) float v2f;
typedef __attribute__((ext_vector_type(8))) float v8f;

#define CDIV(a, b) (((a) + (b) - 1) / (b))

// ---------- helpers ----------
__device__ __forceinline__ unsigned enc_ord(float f) {
    unsigned u = __float_as_uint(f);
    return (u & 0x80000000u) ? ~u : (u | 0x80000000u);
}
__device__ __forceinline__ float dec_ord(unsigned u) {
    unsigned b = (u & 0x80000000u) ? (u & 0x7FFFFFFFu) : ~u;
    return __uint_as_float(b);
}
__device__ __forceinline__ float wave_sum(float v) {
#pragma unroll
    for (int off = 16; off > 0; off >>= 1) v += __shfl_xor(v, off, 32);
    return v;
}

// ---------- generic fill / copy ----------
__global__ void k_fill_f32(float* p, float v, int n) {
    int i = blockIdx.x * 256 + threadIdx.x;
    if (i < n) p[i] = v;
}
__global__ void k_fill_u32(unsigned* p, unsigned v, int n) {
    int i = blockIdx.x * 256 + threadIdx.x;
    if (i < n) p[i] = v;
}
__global__ void k_copy_f32(float* dst, const float* src, int n) {
    int i = blockIdx.x * 256 + threadIdx.x;
    if (i < n) dst[i] = src[i];
}

// ---------- WMMA fp32 GEMM: C[rows x Dout] = act(A[rows x K] @ W[K x Dout] + b) ----------
// One wave per 16x16 output tile; V_WMMA_F32_16X16X4_F32, K stepped by 4.
// A-matrix lane layout (32-bit A 16x4): row = lane&15; a.x = A[row][k+2*half], a.y = A[row][k+1+2*half]
// B-matrix (4x16): b.x = W[k+2*half][col], b.y = W[k+1+2*half][col], col = lane&15
// C/D: VGPR r -> row = r + 8*half, col = lane&15
template <int ACT>
__global__ __launch_bounds__(32) void k_gemm16(const float* A, const float* W,
                                               const float* bias, float* C,
                                               int K, int Dout) {
    const int rowBase = blockIdx.x * 16;
    const int colBase = blockIdx.y * 16;
    const int lane = threadIdx.x;
    const int l = lane & 15;
    const int half = lane >> 4;

    const float* arow = A + (size_t)(rowBase + l) * K + 2 * half;
    const float* wcol = W + colBase + l;

    v8f c = {0.f, 0.f, 0.f, 0.f, 0.f, 0.f, 0.f, 0.f};
    for (int k = 0; k < K; k += 4) {
        v2f a, b;
        a.x = arow[k];
        a.y = arow[k + 1];
        b.x = wcol[(size_t)(k + 2 * half) * Dout];
        b.y = wcol[(size_t)(k + 2 * half + 1) * Dout];
        c = __builtin_amdgcn_wmma_f32_16x16x4_f32(false, a, false, b, (short)0, c,
                                                  false, false);
    }
    const float bv = bias[colBase + l];
    const int row0 = rowBase + 8 * half;
#pragma unroll
    for (int r = 0; r < 8; ++r) {
        float v = c[r] + bv;
        if (ACT == 1) v = fmaxf(v, 0.f);
        C[(size_t)(row0 + r) * Dout + colBase + l] = v;
    }
}

// ---------- GINEConv edge pass: m = relu(x[src] + ea@We + be); h0[dst] += m ----------
// also accumulates per-dst edge_attr sum and count (for self-loop fill 'mean')
__global__ __launch_bounds__(256) void k_gine_edge(const float* x, const int* src,
                                                   const int* dst, const float* ea,
                                                   const float* We, const float* be,
                                                   float* h0, float* loop_s,
                                                   float* cnt, int E) {
    int e = blockIdx.x * 8 + (threadIdx.x >> 5);
    if (e >= E) return;
    int lane = threadIdx.x & 31;
    int s = src[e], d = dst[e];
    float a0 = ea[(size_t)e * 4 + 0], a1 = ea[(size_t)e * 4 + 1];
    float a2 = ea[(size_t)e * 4 + 2], a3 = ea[(size_t)e * 4 + 3];
    int f = lane;  // 32 features == 32 lanes
    float m = x[(size_t)s * 32 + f] + a0 * We[f] + a1 * We[32 + f] +
              a2 * We[64 + f] + a3 * We[96 + f] + be[f];
    m = fmaxf(m, 0.f);
    atomicAdd(&h0[(size_t)d * 32 + f], m);
    if (lane < 4) atomicAdd(&loop_s[(size_t)d * 4 + lane], ea[(size_t)e * 4 + lane]);
    if (lane == 0) atomicAdd(&cnt[d], 1.f);
}

__global__ void k_loop_attr(const float* loop_s, const float* cnt, float* loop_attr,
                            int N) {
    int i = blockIdx.x * 256 + threadIdx.x;
    if (i >= N * 4) return;
    int n = i >> 2;
    loop_attr[i] = loop_s[i] / fmaxf(cnt[n], 1.f);
}

// ---------- GATv2 pass 1: per-(edge,head) score + segment max ----------
__global__ __launch_bounds__(256) void k_gat_score(const float* xl, const float* xr,
                                                   const int* src, const int* dst,
                                                   const float* ea,
                                                   const float* loop_attr,
                                                   const float* We, const float* att,
                                                   float* score, unsigned* m_enc,
                                                   int E, int E2, int H, int CH) {
    int idx = blockIdx.x * 8 + (threadIdx.x >> 5);  // over E2*H
    int e = idx / H;
    if (e >= E2) return;
    int h = idx - e * H;
    int lane = threadIdx.x & 31;
    int s, d;
    const float* eap;
    if (e < E) {
        s = src[e]; d = dst[e]; eap = ea + (size_t)e * 4;
    } else {
        s = e - E; d = s; eap = loop_attr + (size_t)(e - E) * 4;
    }
    float a0 = eap[0], a1 = eap[1], a2 = eap[2], a3 = eap[3];
    int D = H * CH;
    float acc = 0.f;
    for (int c = lane; c < CH; c += 32) {
        int j = h * CH + c;
        float ew = a0 * We[j] + a1 * We[D + j] + a2 * We[2 * D + j] + a3 * We[3 * D + j];
        float v = xl[(size_t)s * D + j] + xr[(size_t)d * D + j] + ew;
        v = (v > 0.f) ? v : 0.2f * v;  // leaky_relu(0.2)
        acc += att[h * CH + c] * v;
    }
    acc = wave_sum(acc);
    if (lane == 0) {
        score[(size_t)e * H + h] = acc;
        atomicMax(&m_enc[(size_t)d * H + h], enc_ord(acc));
    }
}

// ---------- GATv2 pass 2: ex = exp(score - max); z[dst] += ex; acc[dst] += xl[src]*ex ----------
__global__ __launch_bounds__(256) void k_gat_accum(const float* xl, const int* src,
                                                   const int* dst, const float* score,
                                                   const unsigned* m_enc, float* z,
                                                   float* acc, int E, int E2, int H,
                                                   int CH) {
    int e = blockIdx.x * 8 + (threadIdx.x >> 5);
    if (e >= E2) return;
    int lane = threadIdx.x & 31;
    int s, d;
    if (e < E) { s = src[e]; d = dst[e]; } else { s = e - E; d = s; }
    int D = H * CH;
    float ex[4];
    for (int h = 0; h < H; ++h) {
        float m = dec_ord(m_enc[(size_t)d * H + h]);
        ex[h] = expf(score[(size_t)e * H + h] - m);
    }
    if (lane < H) atomicAdd(&z[(size_t)d * H + lane], ex[lane]);
    for (int j = lane; j < D; j += 32)
        atomicAdd(&acc[(size_t)d * D + j], xl[(size_t)s * D + j] * ex[j / CH]);
}

// ---------- GATv2 epilogues ----------
__global__ void k_gat_node_concat(const float* acc, const float* z, const float* bias,
                                  float* out, int N, int H, int CH) {
    int D = H * CH;
    int i = blockIdx.x * 256 + threadIdx.x;
    if (i >= N * D) return;
    int n = i / D, j = i - n * D;
    float v = acc[i] / z[(size_t)n * H + j / CH] + bias[j];
    out[i] = fmaxf(v, 0.f);
}
__global__ void k_gat_node_mean2(const float* acc, const float* z, const float* bias,
                                 float* out, int N, int CH) {
    int i = blockIdx.x * 256 + threadIdx.x;
    if (i >= N * CH) return;
    int n = i / CH, c = i - n * CH;
    float v0 = acc[(size_t)n * 2 * CH + c] / z[(size_t)n * 2 + 0];
    float v1 = acc[(size_t)n * 2 * CH + CH + c] / z[(size_t)n * 2 + 1];
    out[i] = fmaxf(0.5f * (v0 + v1) + bias[c], 0.f);
}

// ---------- attentional pooling ----------
__global__ __launch_bounds__(256) void k_pool_gate(const float* x3, const float* pW,
                                                   const float* pb, const int* batch,
                                                   float* gate, unsigned* gm, int N) {
    int n = blockIdx.x * 8 + (threadIdx.x >> 5);
    if (n >= N) return;
    int lane = threadIdx.x & 31;
    float acc = 0.f;
    for (int c = lane; c < 128; c += 32) acc += x3[(size_t)n * 128 + c] * pW[c];
    acc = wave_sum(acc);
    if (lane == 0) {
        float g = 1.f / (1.f + expf(-(acc + pb[0])));
        gate[n] = g;
        atomicMax(&gm[batch[n]], enc_ord(g));
    }
}
__global__ __launch_bounds__(256) void k_pool_accum(const float* x3, const float* gate,
                                                    const unsigned* gm, const int* batch,
                                                    float* gz, float* emb, int N) {
    int n = blockIdx.x * 8 + (threadIdx.x >> 5);
    if (n >= N) return;
    int lane = threadIdx.x & 31;
    int b = batch[n];
    float ex = expf(gate[n] - dec_ord(gm[b]));
    if (lane == 0) atomicAdd(&gz[b], ex);
    for (int c = lane; c < 128; c += 32)
        atomicAdd(&emb[(size_t)b * 128 + c], ex * x3[(size_t)n * 128 + c]);
}
__global__ void k_pool_norm(float* emb, const float* gz, int G) {
    int i = blockIdx.x * 256 + threadIdx.x;
    if (i >= G * 128) return;
    float zz = gz[i >> 7];
    emb[i] = (zz > 0.f) ? emb[i] / zz : 0.f;
}

// ---------- heads ----------
__global__ void k_mask(float* out, const unsigned char* mask, int n) {
    int i = blockIdx.x * 256 + threadIdx.x;
    if (i >= n) return;
    if (!mask[i]) out[i] = -3.402823466e38f;
}
__global__ __launch_bounds__(256) void k_val(const float* hid, const float* W,
                                             const float* b, float* out, int G) {
    int g = blockIdx.x * 8 + (threadIdx.x >> 5);
    if (g >= G) return;
    int lane = threadIdx.x & 31;
    float acc = 0.f;
    for (int c = lane; c < 64; c += 32) acc += hid[(size_t)g * 64 + c] * W[c];
    acc = wave_sum(acc);
    if (lane == 0) out[g] = acc + b[0];
}

// =====================================================================
extern "C" void kernel_launch(void* const* d_in, const int* in_sizes, int n_in,
                              void* d_out, int out_size, void* d_ws, size_t ws_size,
                              hipStream_t stream) {
    const float* x      = (const float*)d_in[0];
    const int* eidx     = (const int*)d_in[1];
    const float* eattr  = (const float*)d_in[2];
    const int* batch    = (const int*)d_in[3];
    const unsigned char* amask = (const unsigned char*)d_in[4];
    const float* gin_We = (const float*)d_in[6];
    const float* gin_be = (const float*)d_in[7];
    const float* gin_W1 = (const float*)d_in[8];
    const float* gin_b1 = (const float*)d_in[9];
    const float* gin_W2 = (const float*)d_in[10];
    const float* gin_b2 = (const float*)d_in[11];
    const float* g2_Wl  = (const float*)d_in[12];
    const float* g2_bl  = (const float*)d_in[13];
    const float* g2_Wr  = (const float*)d_in[14];
    const float* g2_br  = (const float*)d_in[15];
    const float* g2_We  = (const float*)d_in[16];
    const float* g2_att = (const float*)d_in[17];
    const float* g2_bias= (const float*)d_in[18];
    const float* g3_Wl  = (const float*)d_in[19];
    const float* g3_bl  = (const float*)d_in[20];
    const float* g3_Wr  = (const float*)d_in[21];
    const float* g3_br  = (const float*)d_in[22];
    const float* g3_We  = (const float*)d_in[23];
    const float* g3_att = (const float*)d_in[24];
    const float* g3_bias= (const float*)d_in[25];
    const float* pool_W = (const float*)d_in[26];
    const float* pool_b = (const float*)d_in[27];
    const float* act_W1 = (const float*)d_in[28];
    const float* act_b1 = (const float*)d_in[29];
    const float* act_W2 = (const float*)d_in[30];
    const float* act_b2 = (const float*)d_in[31];
    const float* cr_W1  = (const float*)d_in[32];
    const float* cr_b1  = (const float*)d_in[33];
    const float* cr_W2  = (const float*)d_in[34];
    const float* cr_b2  = (const float*)d_in[35];

    const int N = in_sizes[0] / 32;
    const int E = in_sizes[1] / 2;
    const int G = in_sizes[4] / 80;
    const int E2 = E + N;
    const int* srcp = eidx;
    const int* dstp = eidx + E;

    // workspace carve (bump allocator)
    uint8_t* w = (uint8_t*)d_ws;
    auto alloc = [&](size_t b) -> void* {
        void* p = (void*)w;
        w += (b + 255) & ~(size_t)255;
        return p;
    };
    float* xl       = (float*)alloc((size_t)N * 256 * 4);
    float* xr       = (float*)alloc((size_t)N * 256 * 4);
    float* acc      = (float*)alloc((size_t)N * 256 * 4);  // GAT2 acc -> x2 (in place) -> GAT3 acc
    float* h0       = (float*)alloc((size_t)N * 32 * 4);
    float* t64      = (float*)alloc((size_t)N * 64 * 4);
    float* x1       = (float*)alloc((size_t)N * 64 * 4);
    float* x3       = (float*)alloc((size_t)N * 128 * 4);
    float* loop_s   = (float*)alloc((size_t)N * 4 * 4);
    float* loop_at  = (float*)alloc((size_t)N * 4 * 4);
    float* cnt      = (float*)alloc((size_t)N * 4);
    float* score    = (float*)alloc((size_t)E2 * 4 * 4);
    unsigned* m_enc = (unsigned*)alloc((size_t)N * 4 * 4);
    float* zbuf     = (float*)alloc((size_t)N * 4 * 4);
    float* gate     = (float*)alloc((size_t)N * 4);
    unsigned* gm    = (unsigned*)alloc((size_t)G * 4);
    float* gz       = (float*)alloc((size_t)G * 4);
    float* emb      = (float*)alloc((size_t)G * 128 * 4);
    float* hidA     = (float*)alloc((size_t)G * 64 * 4);
    float* hidC     = (float*)alloc((size_t)G * 64 * 4);

    const unsigned ENC_NEG_MAX = 0x00800000u;  // enc_ord(-FLT_MAX)

    // ---- GINEConv ----
    k_copy_f32<<<CDIV(N * 32, 256), 256, 0, stream>>>(h0, x, N * 32);
    k_fill_f32<<<CDIV(N * 4, 256), 256, 0, stream>>>(loop_s, 0.f, N * 4);
    k_fill_f32<<<CDIV(N, 256), 256, 0, stream>>>(cnt, 0.f, N);
    k_gine_edge<<<CDIV(E, 8), 256, 0, stream>>>(x, srcp, dstp, eattr, gin_We, gin_be,
                                                h0, loop_s, cnt, E);
    k_loop_attr<<<CDIV(N * 4, 256), 256, 0, stream>>>(loop_s, cnt, loop_at, N);
    k_gemm16<1><<<dim3(N / 16, 4), 32, 0, stream>>>(h0, gin_W1, gin_b1, t64, 32, 64);
    k_gemm16<1><<<dim3(N / 16, 4), 32, 0, stream>>>(t64, gin_W2, gin_b2, x1, 64, 64);

    // ---- GATv2 layer 2 (H=4, CH=64, concat) ----
    k_gemm16<0><<<dim3(N / 16, 16), 32, 0, stream>>>(x1, g2_Wl, g2_bl, xl, 64, 256);
    k_gemm16<0><<<dim3(N / 16, 16), 32, 0, stream>>>(x1, g2_Wr, g2_br, xr, 64, 256);
    k_fill_u32<<<CDIV(N * 4, 256), 256, 0, stream>>>(m_enc, ENC_NEG_MAX, N * 4);
    k_fill_f32<<<CDIV(N * 4, 256), 256, 0, stream>>>(zbuf, 0.f, N * 4);
    k_fill_f32<<<CDIV(N * 256, 256), 256, 0, stream>>>(acc, 0.f, N * 256);
    k_gat_score<<<CDIV(E2 * 4, 8), 256, 0, stream>>>(xl, xr, srcp, dstp, eattr, loop_at,
                                                     g2_We, g2_att, score, m_enc, E, E2,
                                                     4, 64);
    k_gat_accum<<<CDIV(E2, 8), 256, 0, stream>>>(xl, srcp, dstp, score, m_enc, zbuf,
                                                 acc, E, E2, 4, 64);
    k_gat_node_concat<<<CDIV(N * 256, 256), 256, 0, stream>>>(acc, zbuf, g2_bias, acc,
                                                              N, 4, 64);  // x2 in-place

    // ---- GATv2 layer 3 (H=2, CH=128, mean) ----  (x2 lives in `acc`)
    k_gemm16<0><<<dim3(N / 16, 16), 32, 0, stream>>>(acc, g3_Wl, g3_bl, xl, 256, 256);
    k_gemm16<0><<<dim3(N / 16, 16), 32, 0, stream>>>(acc, g3_Wr, g3_br, xr, 256, 256);
    k_fill_u32<<<CDIV(N * 2, 256), 256, 0, stream>>>(m_enc, ENC_NEG_MAX, N * 2);
    k_fill_f32<<<CDIV(N * 2, 256), 256, 0, stream>>>(zbuf, 0.f, N * 2);
    k_fill_f32<<<CDIV(N * 256, 256), 256, 0, stream>>>(acc, 0.f, N * 256);
    k_gat_score<<<CDIV(E2 * 2, 8), 256, 0, stream>>>(xl, xr, srcp, dstp, eattr, loop_at,
                                                     g3_We, g3_att, score, m_enc, E, E2,
                                                     2, 128);
    k_gat_accum<<<CDIV(E2, 8), 256, 0, stream>>>(xl, srcp, dstp, score, m_enc, zbuf,
                                                 acc, E, E2, 2, 128);
    k_gat_node_mean2<<<CDIV(N * 128, 256), 256, 0, stream>>>(acc, zbuf, g3_bias, x3, N,
                                                             128);

    // ---- attentional pooling ----
    k_fill_u32<<<CDIV(G, 256), 256, 0, stream>>>(gm, ENC_NEG_MAX, G);
    k_fill_f32<<<CDIV(G, 256), 256, 0, stream>>>(gz, 0.f, G);
    k_fill_f32<<<CDIV(G * 128, 256), 256, 0, stream>>>(emb, 0.f, G * 128);
    k_pool_gate<<<CDIV(N, 8), 256, 0, stream>>>(x3, pool_W, pool_b, batch, gate, gm, N);
    k_pool_accum<<<CDIV(N, 8), 256, 0, stream>>>(x3, gate, gm, batch, gz, emb, N);
    k_pool_norm<<<CDIV(G * 128, 256), 256, 0, stream>>>(emb, gz, G);

    // ---- heads ----
    float* logits = (float*)d_out;
    float* val    = (float*)d_out + (size_t)G * 80;
    k_gemm16<1><<<dim3(G / 16, 4), 32, 0, stream>>>(emb, act_W1, act_b1, hidA, 128, 64);
    k_gemm16<0><<<dim3(G / 16, 5), 32, 0, stream>>>(hidA, act_W2, act_b2, logits, 64, 80);
    k_mask<<<CDIV(G * 80, 256), 256, 0, stream>>>(logits, amask, G * 80);
    k_gemm16<1><<<dim3(G / 16, 4), 32, 0, stream>>>(emb, cr_W1, cr_b1, hidC, 128, 64);
    k_val<<<CDIV(G, 8), 256, 0, stream>>>(hidC, cr_W2, cr_b2, val, G);
}